// Net_61521111548294
// MI455X (gfx1250) — compile-verified
//
#include <hip/hip_runtime.h>
#include <hip/hip_bf16.h>
#include <math.h>

#define N_NODES 100000
#define N_EDGES 3200000
#define N_FEAT  512
#define N_HID   16
#define N_CLS   7

typedef __attribute__((ext_vector_type(2))) float v2f;
typedef __attribute__((ext_vector_type(8))) float v8f;

// ---------------------------------------------------------------------------
// Zero a float buffer
// ---------------------------------------------------------------------------
__global__ void zero_kernel(float* __restrict__ p, int n) {
    int i = blockIdx.x * blockDim.x + threadIdx.x;
    if (i < n) p[i] = 0.0f;
}

// ---------------------------------------------------------------------------
// GEMM1: h[N_NODES,16] = x[N_NODES,512] @ W1[512,16]   (FP32 WMMA 16x16x4)
// 8 waves per block, each wave computes one 16-row tile. W1 staged in LDS.
// ---------------------------------------------------------------------------
__global__ void gemm1_wmma(const float* __restrict__ x,
                           const float* __restrict__ W1,
                           float* __restrict__ h) {
    __shared__ float Wlds[N_FEAT * N_HID];   // 32 KB
    const int tid = threadIdx.x;
    // cooperative load of W1 into LDS (row-major [512,16])
    for (int i = tid; i < N_FEAT * N_HID; i += 256) Wlds[i] = W1[i];
    __syncthreads();

    const int wave = tid >> 5;          // 0..7
    const int lane = tid & 31;
    const int half = lane >> 4;         // 0: K lo pair, 1: K hi pair
    const int sub  = lane & 15;

    const int rowbase = (blockIdx.x * 8 + wave) * 16;
    if (rowbase >= N_NODES) return;     // wave-uniform

    // clamp row for loads (stores are predicated per-lane later)
    int m = rowbase + sub;
    if (m >= N_NODES) m = N_NODES - 1;
    const float* xrow = x + (long)m * N_FEAT;

    v8f c = {};
    for (int k = 0; k < N_FEAT; k += 4) {
        const int kk = k + 2 * half;
        // A fragment 16x4 f32: lane holds x[m][kk], x[m][kk+1]  (8B aligned)
        v2f a = *(const v2f*)(xrow + kk);
        // B fragment 4x16 f32: lane holds W1[kk][n], W1[kk+1][n]
        v2f b;
        b.x = Wlds[kk * N_HID + sub];
        b.y = Wlds[(kk + 1) * N_HID + sub];
        c = __builtin_amdgcn_wmma_f32_16x16x4_f32(
                false, a, false, b, (short)0, c, false, false);
    }

    // C/D layout: VGPR r -> M = r + 8*half, N = sub
    #pragma unroll
    for (int r = 0; r < 8; ++r) {
        int mr = rowbase + r + 8 * half;
        if (mr < N_NODES) h[mr * N_HID + sub] = c[r];
    }
}

// ---------------------------------------------------------------------------
// Scatter-add layer 1: agg1[dst] += h[src]   (16 feats, 4 per thread)
// ---------------------------------------------------------------------------
__global__ void scatter1_kernel(const int* __restrict__ edges,
                                const float* __restrict__ h,
                                float* __restrict__ agg1) {
    long t = (long)blockIdx.x * blockDim.x + threadIdx.x;
    if (t >= (long)N_EDGES * 4) return;
    int e   = (int)(t >> 2);
    int sub = (int)(t & 3);
    int s = edges[e];
    int d = edges[N_EDGES + e];
    float4 v = *(const float4*)(h + (long)s * N_HID + sub * 4);
    float* base = agg1 + (long)d * N_HID + sub * 4;
    atomicAdd(base + 0, v.x);
    atomicAdd(base + 1, v.y);
    atomicAdd(base + 2, v.z);
    atomicAdd(base + 3, v.w);
}

// ---------------------------------------------------------------------------
// h1 = relu(agg1 + b1)   (in a separate buffer slot: done in place on agg1)
// ---------------------------------------------------------------------------
__global__ void relu_bias_kernel(float* __restrict__ agg1,
                                 const float* __restrict__ b1) {
    int i = blockIdx.x * blockDim.x + threadIdx.x;
    if (i < N_NODES * N_HID) {
        float v = agg1[i] + b1[i & 15];
        agg1[i] = v > 0.0f ? v : 0.0f;
    }
}

// ---------------------------------------------------------------------------
// GEMM2: h2[N_NODES,16] = h1[N_NODES,16] @ W2pad[16,16]  (FP32 WMMA 16x16x4)
// W2 is [16,7], zero-padded to 16 columns in LDS. Only cols 0..6 meaningful.
// ---------------------------------------------------------------------------
__global__ void gemm2_wmma(const float* __restrict__ h1,
                           const float* __restrict__ W2,
                           float* __restrict__ h2) {
    __shared__ float Wlds[16 * 16];
    const int tid = threadIdx.x;
    if (tid < 256) {
        int row = tid >> 4, col = tid & 15;
        Wlds[tid] = (col < N_CLS) ? W2[row * N_CLS + col] : 0.0f;
    }
    __syncthreads();

    const int wave = tid >> 5;
    const int lane = tid & 31;
    const int half = lane >> 4;
    const int sub  = lane & 15;

    const int rowbase = (blockIdx.x * 8 + wave) * 16;
    if (rowbase >= N_NODES) return;     // wave-uniform

    int m = rowbase + sub;
    if (m >= N_NODES) m = N_NODES - 1;
    const float* arow = h1 + (long)m * N_HID;

    v8f c = {};
    #pragma unroll
    for (int k = 0; k < N_HID; k += 4) {
        const int kk = k + 2 * half;
        v2f a = *(const v2f*)(arow + kk);
        v2f b;
        b.x = Wlds[kk * 16 + sub];
        b.y = Wlds[(kk + 1) * 16 + sub];
        c = __builtin_amdgcn_wmma_f32_16x16x4_f32(
                false, a, false, b, (short)0, c, false, false);
    }

    #pragma unroll
    for (int r = 0; r < 8; ++r) {
        int mr = rowbase + r + 8 * half;
        if (mr < N_NODES) h2[mr * 16 + sub] = c[r];
    }
}

// ---------------------------------------------------------------------------
// Scatter-add layer 2: agg2[dst] += h2[src]  (7 feats, stored in stride-8 rows)
// 2 threads per edge, each covers a float4 quad (feat 7 ignored).
// ---------------------------------------------------------------------------
__global__ void scatter2_kernel(const int* __restrict__ edges,
                                const float* __restrict__ h2,
                                float* __restrict__ agg2) {
    long t = (long)blockIdx.x * blockDim.x + threadIdx.x;
    if (t >= (long)N_EDGES * 2) return;
    int e   = (int)(t >> 1);
    int sub = (int)(t & 1);
    int s = edges[e];
    int d = edges[N_EDGES + e];
    float4 v = *(const float4*)(h2 + (long)s * 16 + sub * 4);
    float* base = agg2 + (long)d * 8 + sub * 4;
    #pragma unroll
    for (int j = 0; j < 4; ++j) {
        int feat = sub * 4 + j;
        if (feat < N_CLS) {
            float val = (j == 0) ? v.x : (j == 1) ? v.y : (j == 2) ? v.z : v.w;
            atomicAdd(base + j, val);
        }
    }
}

// ---------------------------------------------------------------------------
// out = log_softmax(agg2 + b2)  per node (7 classes)
// ---------------------------------------------------------------------------
__global__ void logsoftmax_kernel(const float* __restrict__ agg2,
                                  const float* __restrict__ b2,
                                  float* __restrict__ out) {
    int n = blockIdx.x * blockDim.x + threadIdx.x;
    if (n >= N_NODES) return;
    float o[N_CLS];
    float mx = -INFINITY;
    #pragma unroll
    for (int j = 0; j < N_CLS; ++j) {
        o[j] = agg2[(long)n * 8 + j] + b2[j];
        mx = fmaxf(mx, o[j]);
    }
    float sum = 0.0f;
    #pragma unroll
    for (int j = 0; j < N_CLS; ++j) sum += __expf(o[j] - mx);
    float lse = mx + __logf(sum);
    #pragma unroll
    for (int j = 0; j < N_CLS; ++j) out[(long)n * N_CLS + j] = o[j] - lse;
}

// ---------------------------------------------------------------------------
// Launch
// ---------------------------------------------------------------------------
extern "C" void kernel_launch(void* const* d_in, const int* in_sizes, int n_in,
                              void* d_out, int out_size, void* d_ws, size_t ws_size,
                              hipStream_t stream) {
    const float* x     = (const float*)d_in[0];
    const int*   edges = (const int*)d_in[1];   // [2, N_EDGES] int32
    const float* W1    = (const float*)d_in[2];
    const float* b1    = (const float*)d_in[3];
    const float* W2    = (const float*)d_in[4];
    const float* b2    = (const float*)d_in[5];
    float* out = (float*)d_out;

    // workspace layout (bytes)
    char* ws = (char*)d_ws;
    float* h    = (float*)(ws);                       // 100000*16 f32 = 6.4 MB
    float* agg1 = (float*)(ws + 6400000);             // 6.4 MB (becomes h1 after relu)
    float* h2   = (float*)(ws + 12800000);            // 6.4 MB
    float* agg2 = (float*)(ws + 19200000);            // 100000*8 f32 = 3.2 MB

    // zero accumulators
    zero_kernel<<<(N_NODES * N_HID + 255) / 256, 256, 0, stream>>>(agg1, N_NODES * N_HID);
    zero_kernel<<<(N_NODES * 8 + 255) / 256, 256, 0, stream>>>(agg2, N_NODES * 8);

    // layer 1
    const int tiles_blocks = (N_NODES + 127) / 128;   // 128 rows per block (8 waves x 16)
    gemm1_wmma<<<tiles_blocks, 256, 0, stream>>>(x, W1, h);
    {
        long total = (long)N_EDGES * 4;
        scatter1_kernel<<<(int)((total + 255) / 256), 256, 0, stream>>>(edges, h, agg1);
    }
    relu_bias_kernel<<<(N_NODES * N_HID + 255) / 256, 256, 0, stream>>>(agg1, b1);

    // layer 2
    gemm2_wmma<<<tiles_blocks, 256, 0, stream>>>(agg1, W2, h2);
    {
        long total = (long)N_EDGES * 2;
        scatter2_kernel<<<(int)((total + 255) / 256), 256, 0, stream>>>(edges, h2, agg2);
    }

    // epilogue
    logsoftmax_kernel<<<(N_NODES + 255) / 256, 256, 0, stream>>>(agg2, b2, out);
}